// RSPDNet_8426725834946
// MI455X (gfx1250) — compile-verified
//
#include <hip/hip_runtime.h>
#include <hip/hip_bf16.h>
#include <math.h>

#define EPS 0.0001f

typedef __attribute__((ext_vector_type(2))) float        v2f;
typedef __attribute__((ext_vector_type(8))) float        v8f;
typedef __attribute__((ext_vector_type(4))) unsigned int u32x4;
typedef __attribute__((ext_vector_type(8))) int          i32x8;
typedef __attribute__((ext_vector_type(4))) int          i32x4;

// ---------------------------------------------------------------------------
// Vectorized LDS zero fill (ds_store_b128). nfloats must be a multiple of 4
// and the buffer 16-byte aligned.
// ---------------------------------------------------------------------------
__device__ __forceinline__ void lds_zero4(float* p, int nfloats, int lane) {
  float4* p4 = (float4*)p;
  const int n4 = nfloats >> 2;
  const float4 z = make_float4(0.f, 0.f, 0.f, 0.f);
  for (int i = lane; i < n4; i += 32) p4[i] = z;
}

// ---------------------------------------------------------------------------
// Unconditional fragment fetch: all operand buffers are zero-padded in LDS so
// no bounds checks (and no EXEC-mask churn) are needed. Garbage on one side
// of a MAC is always multiplied by a guaranteed zero on the other side.
// ---------------------------------------------------------------------------
__device__ __forceinline__ float fetch_elem(const float* P, int ld, bool trans,
                                            int r, int c) {
  return trans ? P[c * ld + r] : P[r * ld + c];
}

// ---------------------------------------------------------------------------
// C(MxN) = A(MxK) * B(KxN) on the f32 WMMA pipe (V_WMMA_F32_16X16X4_F32).
// A fragment (16x4 f32): lanes 0-15 -> M=lane, K={0,1}; lanes 16-31 -> K={2,3}
// B fragment (4x16 f32): VGPR0 rows {0,1} (lane halves), VGPR1 rows {2,3}
// C/D (16x16 f32): VGPR r: lanes 0-15 -> (M=r, N=lane); 16-31 -> (M=8+r)
// All buffers padded to 16-multiples with zeros; fetches are unconditional.
// ---------------------------------------------------------------------------
__device__ __forceinline__ void wmma_gemm(const float* A, int lda, bool tA,
                                          const float* B, int ldb, bool tB,
                                          float* Cd, int ldc,
                                          int M, int N, int K, int lane)
{
  const int mt_n = (M + 15) >> 4;
  const int nt_n = (N + 15) >> 4;
  const int kt_n = (K + 3) >> 2;
  const int half = lane >> 4;   // 0 or 1
  const int l15  = lane & 15;
  for (int mt = 0; mt < mt_n; ++mt) {
    for (int nt = 0; nt < nt_n; ++nt) {
      v8f acc = {0.f, 0.f, 0.f, 0.f, 0.f, 0.f, 0.f, 0.f};
      for (int kt = 0; kt < kt_n; ++kt) {
        v2f a, b;
        const int am = mt * 16 + l15;
        const int ak = kt * 4 + half * 2;
        a.x = fetch_elem(A, lda, tA, am, ak);
        a.y = fetch_elem(A, lda, tA, am, ak + 1);
        const int bn = nt * 16 + l15;
        b.x = fetch_elem(B, ldb, tB, kt * 4 + half,     bn);
        b.y = fetch_elem(B, ldb, tB, kt * 4 + 2 + half, bn);
        acc = __builtin_amdgcn_wmma_f32_16x16x4_f32(
            /*neg_a=*/false, a, /*neg_b=*/false, b,
            /*c_mod=*/(short)0, acc, /*reuse_a=*/false, /*reuse_b=*/false);
      }
#pragma unroll
      for (int r = 0; r < 8; ++r) {
        const int row = mt * 16 + half * 8 + r;
        Cd[row * ldc + nt * 16 + l15] = acc[r];
      }
    }
  }
}

// ---------------------------------------------------------------------------
// One-wave cyclic Jacobi eigensolver for an n x n symmetric matrix in LDS
// (ld=32). Lanes own columns. On exit: diag(A) = eigenvalues, V = eigvecs
// (V fully zero-padded outside n x n).
// ---------------------------------------------------------------------------
__device__ __forceinline__ void jacobi_eig(float* A, float* V, int n,
                                           int sweeps, int lane)
{
  lds_zero4(V, 32 * 32, lane);
  __syncthreads();
  if (lane < n) V[lane * 32 + lane] = 1.0f;
  __syncthreads();
  for (int sw = 0; sw < sweeps; ++sw) {
    for (int p = 0; p < n - 1; ++p) {
      for (int q = p + 1; q < n; ++q) {
        const float apq = A[p * 32 + q];
        const float app = A[p * 32 + p];
        const float aqq = A[q * 32 + q];
        float c, sn;
        if (fabsf(apq) > 1e-12f) {
          const float theta = (aqq - app) / (2.0f * apq);
          const float t = copysignf(1.0f, theta) /
                          (fabsf(theta) + sqrtf(1.0f + theta * theta));
          c  = rsqrtf(1.0f + t * t);
          sn = t * c;
        } else { c = 1.0f; sn = 0.0f; }
        // column update A <- A*J, V <- V*J (lane k owns row k)
        if (lane < n) {
          const float akp = A[lane * 32 + p], akq = A[lane * 32 + q];
          A[lane * 32 + p] = c * akp - sn * akq;
          A[lane * 32 + q] = sn * akp + c * akq;
          const float vkp = V[lane * 32 + p], vkq = V[lane * 32 + q];
          V[lane * 32 + p] = c * vkp - sn * vkq;
          V[lane * 32 + q] = sn * vkp + c * vkq;
        }
        __syncthreads();
        // row update A <- J^T * A (lane k owns column k)
        if (lane < n) {
          const float apk = A[p * 32 + lane], aqk = A[q * 32 + lane];
          A[p * 32 + lane] = c * apk - sn * aqk;
          A[q * 32 + lane] = sn * apk + c * aqk;
        }
        __syncthreads();
      }
    }
  }
}

// ---------------------------------------------------------------------------
// Analytic 2x2 symmetric eigendecomposition + spectral reconstruction.
// ---------------------------------------------------------------------------
__device__ __forceinline__ void sym2_eig(float a, float b, float d,
                                         float& w0, float& w1,
                                         float& c, float& s) {
  const float ht = 0.5f * (a + d);
  const float hd = 0.5f * (a - d);
  const float r  = sqrtf(hd * hd + b * b);
  w0 = ht - r; w1 = ht + r;
  float vx, vy;
  if (r > 1e-20f) {
    if (fabsf(b) > 1e-20f) { vx = b;   vy = w1 - a; }
    else if (a >= d)       { vx = 1.f; vy = 0.f;    }
    else                   { vx = 0.f; vy = 1.f;    }
  } else { vx = 1.f; vy = 0.f; }
  const float inv = rsqrtf(vx * vx + vy * vy);
  c = vx * inv; s = vy * inv;
}

__device__ __forceinline__ void sym2_recon(float f0, float f1, float c, float s,
                                           float& o00, float& o01, float& o11) {
  o00 = f1 * c * c + f0 * s * s;
  o01 = (f1 - f0) * c * s;
  o11 = f1 * s * s + f0 * c * c;
}

// ---------------------------------------------------------------------------
// Kernel A: one wave32 per (b,t) matrix. TDM DMA of the 57x57 tile into LDS,
// then the full W1->W2->W3 rect chain entirely in LDS/WMMA.
// Writes a[(t*128+b)*4 .. +3] = rect(W3^T . W2^T . W1^T x W1 . W2 . W3).
// ---------------------------------------------------------------------------
__global__ __launch_bounds__(32)
void rspd_matrices_kernel(const float* __restrict__ x,
                          const float* __restrict__ W1,
                          const float* __restrict__ W2,
                          const float* __restrict__ W3,
                          float* __restrict__ a_out)
{
  // Padded, zero-filled LDS operands (ld = row stride in floats):
  __shared__ __align__(16) float xs[64 * 57 + 16];  // 57x57 tile, ld=57, tail zero
  __shared__ __align__(16) float m1[64 * 32];       // GEMM scratch, ld=32
  __shared__ __align__(16) float m2[32 * 32];
  __shared__ __align__(16) float m3[32 * 32];       // eigenvectors
  __shared__ __align__(16) float m4[32 * 32];
  __shared__ __align__(16) float w1p[64 * 32];      // W1 (57x20) zero-padded
  __shared__ __align__(16) float w2p[32 * 32];      // W2 (20x10) zero-padded
  __shared__ __align__(16) float w3s[32];           // W3 (10x2) + zero tail

  const int lane = threadIdx.x;
  const int bid  = blockIdx.x;                 // = b*64 + t
  const float* xsrc = x + (size_t)bid * (57 * 57);

  // --- issue the Tensor Data Mover DMA first; it writes only xs[0..3248] ---
#if __has_builtin(__builtin_amdgcn_tensor_load_to_lds)
  {
    // Tensor DMA descriptor: 2-D f32 tensor 57x57, tile 57x57, stride 57.
    const unsigned long long ga = (unsigned long long)(const void*)xsrc;
    const unsigned int lds_off = (unsigned int)(unsigned long long)(void*)&xs[0];
    u32x4 g0; i32x8 g1; i32x4 g2, g3; i32x8 g4;
    g0[0] = 1u;                                              // count=1, flags 0
    g0[1] = lds_off;                                         // lds_addr
    g0[2] = (unsigned int)(ga & 0xFFFFFFFFull);              // global_addr lo
    g0[3] = (unsigned int)((ga >> 32) & 0x01FFFFFFull)       // global_addr hi
          | (2u << 30);                                      // type=2 (image)
    g1[0] = (2 << 16);                                       // data_size=4B
    g1[1] = (int)((57u & 0xFFFFu) << 16);                    // tensor_dim0 lo16
    g1[2] = (int)((57u & 0xFFFFu) << 16);                    // dim0 hi=0 | dim1 lo16
    g1[3] = (int)(57u << 16);                                // dim1 hi=0 | tile_dim0
    g1[4] = 57;                                              // tile_dim1 | tile_dim2=0
    g1[5] = 57;                                              // tensor_dim0_stride lo32
    g1[6] = 0;                                               // strides hi
    g1[7] = 0;
    g2[0] = g2[1] = g2[2] = g2[3] = 0;
    g3[0] = g3[1] = g3[2] = g3[3] = 0;
    g4[0] = g4[1] = g4[2] = g4[3] = 0;
    g4[4] = g4[5] = g4[6] = g4[7] = 0;
    __builtin_amdgcn_tensor_load_to_lds(g0, g1, g2, g3, g4, 0);
  }
#else
  for (int i = lane; i < 57 * 57; i += 32) xs[i] = xsrc[i];
#endif

  // --- zero-fill pads (disjoint from the TDM destination region) ---
  for (int i = 57 * 57 + lane; i < 64 * 57 + 16; i += 32) xs[i] = 0.0f;
  lds_zero4(m1,  64 * 32, lane);
  lds_zero4(m2,  32 * 32, lane);
  lds_zero4(m4,  32 * 32, lane);
  lds_zero4(w1p, 64 * 32, lane);
  lds_zero4(w2p, 32 * 32, lane);
  if (lane < 32) w3s[lane] = 0.0f;
  __syncthreads();

  // --- small weights into padded layouts (L2-resident across 8192 blocks) ---
  for (int i = lane; i < 57 * 20; i += 32) w1p[(i / 20) * 32 + (i % 20)] = W1[i];
  for (int i = lane; i < 20 * 10; i += 32) w2p[(i / 10) * 32 + (i % 10)] = W2[i];
  if (lane < 20) w3s[lane] = W3[lane];

#if __has_builtin(__builtin_amdgcn_s_wait_tensorcnt)
  __builtin_amdgcn_s_wait_tensorcnt((short)0);
#endif
  __syncthreads();

  // ---- Stage 1: 57 -> 20 ----
  wmma_gemm(xs, 57, false, w1p, 32, false, m1, 32, 57, 20, 57, lane); // t1 = x*W1
  __syncthreads();
  wmma_gemm(w1p, 32, true, m1, 32, false, m2, 32, 20, 20, 57, lane);  // A = W1^T*t1
  __syncthreads();
  jacobi_eig(m2, m3, 20, 8, lane);
  if (lane < 20) {
    const float wc = fmaxf(m2[lane * 32 + lane], EPS);
    for (int k = 0; k < 20; ++k) m4[k * 32 + lane] = m3[k * 32 + lane] * wc;
  }
  __syncthreads();
  wmma_gemm(m4, 32, false, m3, 32, true, m2, 32, 20, 20, 20, lane);   // R1 = U*V^T
  __syncthreads();

  // ---- Stage 2: 20 -> 10 ----
  wmma_gemm(m2, 32, false, w2p, 32, false, m1, 32, 20, 10, 20, lane); // t2 = R1*W2
  __syncthreads();
  wmma_gemm(w2p, 32, true, m1, 32, false, m4, 32, 10, 10, 20, lane);  // A = W2^T*t2
  __syncthreads();
  jacobi_eig(m4, m3, 10, 6, lane);
  if (lane < 10) {
    const float wc = fmaxf(m4[lane * 32 + lane], EPS);
    for (int k = 0; k < 10; ++k) m2[k * 32 + lane] = m3[k * 32 + lane] * wc;
  }
  __syncthreads();
  // stale m2 entries at k>=10 are annihilated by m3's zero columns (V padded)
  wmma_gemm(m2, 32, false, m3, 32, true, m1, 32, 10, 10, 10, lane);   // R2 = U*V^T
  __syncthreads();

  // ---- Stage 3: 10 -> 2 (scalar, tiny) ----
  if (lane < 20) {                 // t3 (10x2): lane = r*2+c
    const int r = lane >> 1, c = lane & 1;
    float acc = 0.f;
    for (int k = 0; k < 10; ++k) acc += m1[r * 32 + k] * w3s[k * 2 + c];
    m2[lane] = acc;
  }
  __syncthreads();
  if (lane < 4) {                  // z3 (2x2) = W3^T * t3
    const int r = lane >> 1, c = lane & 1;
    float acc = 0.f;
    for (int k = 0; k < 10; ++k) acc += w3s[k * 2 + r] * m2[k * 2 + c];
    m4[lane] = acc;
  }
  __syncthreads();
  if (lane == 0) {                 // analytic rect of the 2x2, store a_t
    const float a = m4[0], b = 0.5f * (m4[1] + m4[2]), d = m4[3];
    float w0, w1, c, s, r00, r01, r11;
    sym2_eig(a, b, d, w0, w1, c, s);
    sym2_recon(fmaxf(w0, EPS), fmaxf(w1, EPS), c, s, r00, r01, r11);
    const int t = bid & 63, bb = bid >> 6;
    float* dst = a_out + ((size_t)(t * 128 + bb)) * 4;
    dst[0] = r00; dst[1] = r01; dst[2] = r01; dst[3] = r11;
  }
}

// ---------------------------------------------------------------------------
// Kernel B: sequential scan over T=64 with analytic 2x2 eig/log; one thread
// per batch element. Writes out[b][t][0..16].
// ---------------------------------------------------------------------------
__global__ __launch_bounds__(128)
void rspd_scan_kernel(const float* __restrict__ a,
                      const float* __restrict__ lin_w,
                      const float* __restrict__ lin_b,
                      const float* __restrict__ alpha,
                      float* __restrict__ out)
{
  const int b = threadIdx.x + blockIdx.x * blockDim.x;
  if (b >= 128) return;
  const float s = 1.0f / (1.0f + __expf(-alpha[0]));
  float h00 = 0.f, h01 = 0.f, h11 = 0.f;
  for (int t = 0; t < 64; ++t) {
    const float* at = a + ((size_t)t * 128 + b) * 4;
    float w0, w1, c, sn, r00, r01, r11;
    sym2_eig(h00, h01, h11, w0, w1, c, sn);
    sym2_recon(fmaxf(w0, EPS), fmaxf(w1, EPS), c, sn, r00, r01, r11);
    const float n00 = s * at[0] + (1.f - s) * r00;
    const float n01 = s * 0.5f * (at[1] + at[2]) + (1.f - s) * r01;
    const float n11 = s * at[3] + (1.f - s) * r11;
    sym2_eig(n00, n01, n11, w0, w1, c, sn);
    float l00, l01, l11;
    sym2_recon(logf(fmaxf(w0, 1e-30f)), logf(fmaxf(w1, 1e-30f)), c, sn,
               l00, l01, l11);
    float* o = out + ((size_t)b * 64 + t) * 17;
    for (int j = 0; j < 17; ++j)
      o[j] = lin_w[j * 3 + 0] * l00 + lin_w[j * 3 + 1] * l01 +
             lin_w[j * 3 + 2] * l11 + lin_b[j];
    h00 = n00; h01 = n01; h11 = n11;
  }
}

extern "C" void kernel_launch(void* const* d_in, const int* in_sizes, int n_in,
                              void* d_out, int out_size, void* d_ws, size_t ws_size,
                              hipStream_t stream) {
  (void)in_sizes; (void)n_in; (void)out_size; (void)ws_size;
  const float* x     = (const float*)d_in[0];
  const float* W1    = (const float*)d_in[1];
  const float* W2    = (const float*)d_in[2];
  const float* W3    = (const float*)d_in[3];
  const float* lin_w = (const float*)d_in[4];
  const float* lin_b = (const float*)d_in[5];
  const float* alpha = (const float*)d_in[6];
  float* out  = (float*)d_out;
  float* a_ws = (float*)d_ws;   // (T=64, B=128, 2, 2) f32 = 128 KB scratch

  rspd_matrices_kernel<<<dim3(128 * 64), dim3(32), 0, stream>>>(
      x, W1, W2, W3, a_ws);
  rspd_scan_kernel<<<dim3(1), dim3(128), 0, stream>>>(
      a_ws, lin_w, lin_b, alpha, out);
}